// EqvPSRFeedForward_51994874085881
// MI455X (gfx1250) — compile-verified
//
#include <hip/hip_runtime.h>
#include <hip/hip_bf16.h>
#include <math.h>

typedef __attribute__((ext_vector_type(2))) float v2f;
typedef __attribute__((ext_vector_type(8))) float v8f;

#define BB 2
#define NN 256
#define CC 32
#define NBASIS 10
#define HH 64
#define MAXR 10.0f
#define GNORM 1.4230852449f

#define KTOT (NN * HH)         // 16384 contraction length per batch
#define JC 8                   // j's per chunk
#define NCHUNK (NN / JC)       // 32 chunks
#define KP (JC * HH + 4)       // padded LDS row stride (516) -> conflict-free A loads

// ---------------------------------------------------------------------------
// Kernel 1: Gt[b][o][j*H+h] = sum_c Wr2[h][o*C+c] * x[b][j][c]   (k contiguous)
// ---------------------------------------------------------------------------
__global__ void __launch_bounds__(256) g_kernel(const float* __restrict__ x,
                                                const float* __restrict__ Wr2,
                                                float* __restrict__ G) {
    int bj = blockIdx.x;              // 0..511
    int b = bj >> 8, j = bj & 255;
    __shared__ float xs[CC];
    int t = threadIdx.x;
    if (t < CC) xs[t] = x[(b * NN + j) * CC + t];
    __syncthreads();
    // 2048 (h,o) outputs, 256 threads -> 8 each; h = lane-contiguous for
    // coalesced writes along the k (=j*64+h) axis of Gt.
    int h  = t & 63;
    int og = t >> 6;                  // 0..3
    for (int q = 0; q < 8; ++q) {
        int o = og * 8 + q;
        const float* w = &Wr2[h * (CC * CC) + o * CC];
        float s = 0.f;
#pragma unroll
        for (int c = 0; c < CC; ++c) s += w[c] * xs[c];
        G[(b * CC + o) * KTOT + j * HH + h] = s;
    }
}

// ---------------------------------------------------------------------------
// Kernel 2: fused rbf -> radial MLP -> WMMA contraction over (j,h)
// grid = (NCHUNK, 16 i-tiles, B), block = 128 (4 waves)
// partial tile per block: [16 m][32 o] floats
// ---------------------------------------------------------------------------
__global__ void __launch_bounds__(128) conv_kernel(const float* __restrict__ xyz,
                                                   const float* __restrict__ Wr1,
                                                   const float* __restrict__ G,
                                                   float* __restrict__ part) {
    int jc = blockIdx.x;   // 0..31
    int it = blockIdx.y;   // 0..15
    int b  = blockIdx.z;   // 0..1
    __shared__ __align__(16) float hdn[16 * KP];   // 33 KB
    __shared__ __align__(16) float red[4 * 512];   // 8 KB
    int t = threadIdx.x;

    // ---- Phase 1: one (i,j) pair per thread ----
    {
        int ii = t >> 3;               // 0..15
        int jj = t & 7;                // 0..7
        int i = it * 16 + ii;
        int j = jc * JC + jj;
        const float* pi = &xyz[(b * NN + i) * 3];
        const float* pj = &xyz[(b * NN + j) * 3];
        float dx = pi[0] - pj[0], dy = pi[1] - pj[1], dz = pi[2] - pj[2];
        float d = sqrtf(dx * dx + dy * dy + dz * dz + 1e-12f);
        const float step  = MAXR / (NBASIS - 1);
        const float sigma = 0.8f * MAXR / (NBASIS - 1);
        float rb[NBASIS];
#pragma unroll
        for (int k = 0; k < NBASIS; ++k) {
            float u = (d - step * k) / sigma;
            rb[k] = __expf(-u * u) * (1.0f / GNORM);
        }
        float* dst = &hdn[ii * KP + jj * HH];
#pragma unroll 4
        for (int h = 0; h < HH; ++h) {
            float s = 0.f;
#pragma unroll
            for (int k = 0; k < NBASIS; ++k) s += rb[k] * Wr1[k * HH + h];
            dst[h] = s / (1.f + __expf(-s));      // swish
        }
    }
    __syncthreads();

    // ---- Phase 2: WMMA, 4 waves split the chunk's K=512 range ----
    int wave  = t >> 5;
    int lane  = t & 31;
    int m     = lane & 15;       // A: row M ; B: col N ; D: col N
    int khalf = lane >> 4;       // K-half select
    v8f acc0 = {}, acc1 = {};
    int kw = wave * 128;
    int kchunk_base = jc * (JC * HH);
    // Gt rows for this lane's B-fragment column (o = m and o = 16+m)
    const float* g0 = &G[(b * CC + m)      * KTOT + kchunk_base];
    const float* g1 = &G[(b * CC + 16 + m) * KTOT + kchunk_base];
    for (int kk = 0; kk < 128; kk += 4) {
        int kl = kw + kk + khalf * 2;                // local k (lane's pair)
        v2f a  = *(const v2f*)&hdn[m * KP + kl];     // A[m][kl], A[m][kl+1]
        v2f b0 = *(const v2f*)(g0 + kl);             // B[kl][n], B[kl+1][n]  (o-tile 0)
        v2f b1 = *(const v2f*)(g1 + kl);             // (o-tile 1)
        acc0 = __builtin_amdgcn_wmma_f32_16x16x4_f32(false, a, false, b0,
                                                     (short)0, acc0, false, false);
        acc1 = __builtin_amdgcn_wmma_f32_16x16x4_f32(false, a, false, b1,
                                                     (short)0, acc1, false, false);
    }
    // stash wave partials, fixed-order in-block reduction (deterministic)
    float* rw = &red[wave * 512];
#pragma unroll
    for (int r = 0; r < 8; ++r) {
        int mm = r + 8 * khalf;
        rw[mm * 32 + m]      = acc0[r];
        rw[mm * 32 + 16 + m] = acc1[r];
    }
    __syncthreads();
    float* out = &part[(((b * 16 + it) * NCHUNK) + jc) * 512];
    for (int e = t; e < 512; e += 128)
        out[e] = red[e] + red[512 + e] + red[1024 + e] + red[1536 + e];
}

// ---------------------------------------------------------------------------
// Kernel 3: reduce partials, abs, mask, pool, normalize, fc3+lrelu, fc2+sigmoid
// grid = (B), block = 256
// ---------------------------------------------------------------------------
__global__ void __launch_bounds__(256) final_kernel(const int* __restrict__ mask,
                                                    const float* __restrict__ part,
                                                    const float* __restrict__ W3,
                                                    const float* __restrict__ b3,
                                                    const float* __restrict__ W2,
                                                    const float* __restrict__ b2,
                                                    float* __restrict__ out) {
    int b = blockIdx.x;
    int t = threadIdx.x;            // thread t == point i
    __shared__ float ys[256 * 32];  // 32 KB
    __shared__ float p[32], pn[32], q[32], stats[2];
    int it = t >> 4, m = t & 15;
    float dmask = (mask[(b * NN + t) * NN + t] != 0) ? 1.0f : 0.0f;
    for (int o = 0; o < 32; ++o) {
        float s = 0.f;
        for (int jcb = 0; jcb < NCHUNK; ++jcb)
            s += part[(((b * 16 + it) * NCHUNK) + jcb) * 512 + m * 32 + o];
        ys[t * 32 + o] = fabsf(s * 0.0625f) * dmask;   // 1/sqrt(256) = 1/16
    }
    __syncthreads();
    if (t < 32) {
        float s = 0.f;
        for (int i = 0; i < 256; ++i) s += ys[i * 32 + t];
        p[t] = s;
    }
    __syncthreads();
    if (t == 0) {
        float mean = 0.f;
        for (int o = 0; o < 32; ++o) mean += p[o];
        mean *= (1.0f / 32.0f);
        float var = 0.f;
        for (int o = 0; o < 32; ++o) { float d0 = p[o] - mean; var += d0 * d0; }
        var *= (1.0f / 31.0f);                        // ddof = 1
        stats[0] = mean;
        stats[1] = sqrtf(var) + 1e-6f;
    }
    __syncthreads();
    if (t < 32) pn[t] = (p[t] - stats[0]) / stats[1];
    __syncthreads();
    if (t < 32) {
        float s = b3[t];
        for (int c = 0; c < 32; ++c) s += pn[c] * W3[c * 32 + t];
        q[t] = (s > 0.f) ? s : 0.01f * s;             // leaky relu
    }
    __syncthreads();
    if (t == 0) {
        float s = b2[0];
        for (int c = 0; c < 32; ++c) s += q[c] * W2[c];
        out[b] = 1.f / (1.f + __expf(-s));
    }
}

// ---------------------------------------------------------------------------
extern "C" void kernel_launch(void* const* d_in, const int* in_sizes, int n_in,
                              void* d_out, int out_size, void* d_ws, size_t ws_size,
                              hipStream_t stream) {
    const float* x   = (const float*)d_in[0];   // [2,256,32]
    const float* xyz = (const float*)d_in[1];   // [2,256,3]
    const int*   msk = (const int*)d_in[2];     // [2,256,256]
    const float* Wr1 = (const float*)d_in[3];   // [10,64]
    const float* Wr2 = (const float*)d_in[4];   // [64,1024]
    const float* W3  = (const float*)d_in[5];   // [32,32]
    const float* b3  = (const float*)d_in[6];   // [32]
    const float* W2  = (const float*)d_in[7];   // [32,1]
    const float* b2  = (const float*)d_in[8];   // [1]
    float* out = (float*)d_out;                 // [2]

    float* G    = (float*)d_ws;                                   // 4 MB (transposed layout)
    float* part = (float*)((char*)d_ws + (size_t)BB * CC * KTOT * sizeof(float)); // 2 MB

    g_kernel<<<dim3(BB * NN), 256, 0, stream>>>(x, Wr2, G);
    conv_kernel<<<dim3(NCHUNK, 16, BB), 128, 0, stream>>>(xyz, Wr1, G, part);
    final_kernel<<<dim3(BB), 256, 0, stream>>>(msk, part, W3, b3, W2, b2, out);
}